// KMeansSoftDTW_67688684585010
// MI455X (gfx1250) — compile-verified
//
#include <hip/hip_runtime.h>

// Problem constants (match reference)
#define NN 512
#define KK 32
#define TT 128
#define CD 16
#define PITCH 130        // 128 + 2 pad: anti-diagonal LDS reads become stride-129 -> conflict-free
#define INFV 1e30f

typedef float v2f __attribute__((ext_vector_type(2)));
typedef float v8f __attribute__((ext_vector_type(8)));

// One workgroup (128 threads = 4 wave32) per (n,k) pair.
// Phase 1: 8x8 grid of 16x16 tiles via V_WMMA_F32_16X16X4_F32 (K=16 -> 4 chained WMMAs).
// Phase 2: DTW anti-diagonal wavefront with 128 threads (thread t owns row t).
__global__ __launch_bounds__(128) void dtw_cost_kernel(const float* __restrict__ X,
                                                       const float* __restrict__ P,
                                                       float* __restrict__ dists) {
    __shared__ float cost[TT * PITCH];  // 66,560 B
    __shared__ float x2s[TT];
    __shared__ float p2s[TT];
    __shared__ float d1s[TT];
    __shared__ float d2s[TT];

    const int bid  = blockIdx.x;
    const int n    = bid / KK;
    const int k    = bid % KK;
    const int tid  = threadIdx.x;
    const int wave = tid >> 5;
    const int lane = tid & 31;

    const float* Xn = X + (size_t)n * TT * CD;
    const float* Pk = P + (size_t)k * TT * CD;

    // ---- Phase 0: squared norms x2[t], p2[s] ----
    {
        const float4* xr = (const float4*)(Xn + tid * CD);
        const float4* pr = (const float4*)(Pk + tid * CD);
        float sx = 0.f, sp = 0.f;
#pragma unroll
        for (int q = 0; q < 4; ++q) {
            float4 a = xr[q];
            float4 b = pr[q];
            sx += a.x * a.x + a.y * a.y + a.z * a.z + a.w * a.w;
            sp += b.x * b.x + b.y * b.y + b.z * b.z + b.w * b.w;
        }
        x2s[tid] = sx;
        p2s[tid] = sp;
    }
    __syncthreads();

    // ---- Phase 1: cost[t][s] = x2[t] + p2[s] - 2 * dot(X[t,:], P[s,:]) via WMMA ----
    // ISA layout, 32-bit A 16x4: lanes 0-15 hold K=0(v0),K=1(v1); lanes 16-31 hold K=2,K=3.
    // B (4x16) symmetric: v0 = K (half?2:0), v1 = K+1, lanes index N.
    const int half  = lane >> 4;     // 0 or 1
    const int l15   = lane & 15;
    const int khalf = half * 2;      // channel pair selector per half-wave

#pragma unroll
    for (int tw = 0; tw < 2; ++tw) {
        const int ti = wave * 2 + tw;            // t-tile index 0..7
        const float* Arow = Xn + (ti * 16 + l15) * CD + khalf;
        v2f a0 = *(const v2f*)(Arow + 0);
        v2f a1 = *(const v2f*)(Arow + 4);
        v2f a2 = *(const v2f*)(Arow + 8);
        v2f a3 = *(const v2f*)(Arow + 12);

        for (int sj = 0; sj < 8; ++sj) {         // s-tile index 0..7
            const float* Brow = Pk + (sj * 16 + l15) * CD + khalf;
            v2f b0 = *(const v2f*)(Brow + 0);
            v2f b1 = *(const v2f*)(Brow + 4);
            v2f b2 = *(const v2f*)(Brow + 8);
            v2f b3 = *(const v2f*)(Brow + 12);

            v8f acc = {0.f, 0.f, 0.f, 0.f, 0.f, 0.f, 0.f, 0.f};
            acc = __builtin_amdgcn_wmma_f32_16x16x4_f32(false, a0, false, b0, (short)0, acc, false, false);
            acc = __builtin_amdgcn_wmma_f32_16x16x4_f32(false, a1, false, b1, (short)0, acc, false, false);
            acc = __builtin_amdgcn_wmma_f32_16x16x4_f32(false, a2, false, b2, (short)0, acc, false, false);
            acc = __builtin_amdgcn_wmma_f32_16x16x4_f32(false, a3, false, b3, (short)0, acc, false, false);

            // C/D layout: VGPR r -> row M = r + 8*half; lane l15 -> column
            const int scol = sj * 16 + l15;
            const float ps = p2s[scol];
#pragma unroll
            for (int r = 0; r < 8; ++r) {
                const int trow = ti * 16 + r + half * 8;
                cost[trow * PITCH + scol] = x2s[trow] + ps - 2.0f * acc[r];
            }
        }
    }

    // ---- Phase 2: DTW wavefront over anti-diagonals ----
    d1s[tid] = INFV;
    d2s[tid] = INFV;
    float my_d1 = INFV;   // register copy of d1s[t] (only written by this thread)
    __syncthreads();

    const int t = tid;
    for (int d = 0; d < 2 * TT - 1; ++d) {
        const int j  = d - t;
        const float up = (t > 0) ? d1s[t - 1] : INFV;   // (t-1, j)   from prev diagonal
        const float dg = (t > 0) ? d2s[t - 1] : INFV;   // (t-1, j-1) from diagonal-2
        float prev = fminf(my_d1, fminf(up, dg));       // my_d1 == (t, j-1)
        if (d == 0 && t == 0) prev = 0.f;
        const float nd = (j >= 0 && j < TT) ? (cost[t * PITCH + j] + prev) : INFV;
        __syncthreads();
        d2s[t] = my_d1;   // new d2 = old d1
        d1s[t] = nd;
        my_d1  = nd;
        __syncthreads();
    }

    if (t == TT - 1) dists[bid] = my_d1;   // D[127][127]
}

// First-occurrence argmin over K per sequence -> int32 indices
__global__ __launch_bounds__(128) void argmin_kernel(const float* __restrict__ dists,
                                                     int* __restrict__ out) {
    const int n = blockIdx.x * blockDim.x + threadIdx.x;
    if (n >= NN) return;
    const float* row = dists + (size_t)n * KK;
    float best = row[0];
    int   bi   = 0;
#pragma unroll
    for (int kk = 1; kk < KK; ++kk) {
        const float v = row[kk];
        if (v < best) { best = v; bi = kk; }
    }
    out[n] = bi;
}

extern "C" void kernel_launch(void* const* d_in, const int* in_sizes, int n_in,
                              void* d_out, int out_size, void* d_ws, size_t ws_size,
                              hipStream_t stream) {
    const float* X = (const float*)d_in[0];   // (N, T, C) f32
    const float* P = (const float*)d_in[1];   // (K, T, C) f32
    float* dists = (float*)d_ws;              // (N, K) f32 scratch, 64 KB
    int*   out   = (int*)d_out;               // (N,) int32 argmin indices

    dtw_cost_kernel<<<NN * KK, 128, 0, stream>>>(X, P, dists);
    argmin_kernel<<<(NN + 127) / 128, 128, 0, stream>>>(dists, out);
}